// EmbedderGNNv2_85555748536461
// MI455X (gfx1250) — compile-verified
//
#include <hip/hip_runtime.h>

#define GNN_N_NODES 50000
#define GNN_N_EDGES 800000
#define GNN_BN_EPS 1e-5f

typedef __attribute__((ext_vector_type(2))) float v2f;
typedef __attribute__((ext_vector_type(8))) float v8f;

// ---------------------------------------------------------------- utilities
__global__ void gnn_zero_f32_kernel(float* __restrict__ p, long long n) {
  long long i = (long long)blockIdx.x * blockDim.x + threadIdx.x;
  if (i < n) p[i] = 0.0f;
}

__global__ void gnn_zero_i32_kernel(int* __restrict__ p, int n) {
  int i = blockIdx.x * blockDim.x + threadIdx.x;
  if (i < n) p[i] = 0;
}

__global__ void gnn_copy_i32_kernel(const int* __restrict__ s,
                                    int* __restrict__ d, int n) {
  int i = blockIdx.x * blockDim.x + threadIdx.x;
  if (i < n) d[i] = s[i];
}

// ------------------------------------------------------------- CSR build
__global__ void gnn_degree_kernel(const int* __restrict__ dst,
                                  int* __restrict__ deg, int E) {
  int e = blockIdx.x * blockDim.x + threadIdx.x;
  if (e < E) atomicAdd(&deg[dst[e]], 1);
}

// Single-workgroup streaming exclusive scan (Hillis-Steele in LDS).
__global__ void gnn_scan_kernel(const int* __restrict__ deg,
                                int* __restrict__ rowptr, int n) {
  __shared__ int smem[1024];
  __shared__ int carry;
  const int t = threadIdx.x;
  if (t == 0) carry = 0;
  __syncthreads();
  for (int base = 0; base < n; base += 1024) {
    int i = base + t;
    int v = (i < n) ? deg[i] : 0;
    smem[t] = v;
    __syncthreads();
    for (int off = 1; off < 1024; off <<= 1) {
      int add = (t >= off) ? smem[t - off] : 0;
      __syncthreads();
      smem[t] += add;
      __syncthreads();
    }
    int incl = smem[t];
    if (i < n) rowptr[i] = carry + (incl - v);
    __syncthreads();
    if (t == 1023) carry += incl;
    __syncthreads();
  }
  if (t == 0) rowptr[n] = carry;
}

__global__ void gnn_bucket_kernel(const int* __restrict__ src,
                                  const int* __restrict__ dst,
                                  int* __restrict__ cursor,
                                  int* __restrict__ srcSorted, int E) {
  int e = blockIdx.x * blockDim.x + threadIdx.x;
  if (e >= E) return;
  int pos = atomicAdd(&cursor[dst[e]], 1);
  srcSorted[pos] = src[e];
}

// ----------------------------------------------------- atomic-free mean agg
// block = one node, blockDim.x = d features; neighbor row indices staged in
// LDS and broadcast; feature reads fully coalesced. Division fused.
__global__ void gnn_gather_mean_kernel(const float* __restrict__ h,
                                       const int* __restrict__ rowptr,
                                       const int* __restrict__ srcSorted,
                                       float* __restrict__ meanb, int d) {
  __shared__ int sidx[64];
  const int node = blockIdx.x;
  const int f = threadIdx.x;
  const int p0 = rowptr[node];
  const int p1 = rowptr[node + 1];
  float s = 0.0f;
  for (int base = p0; base < p1; base += 64) {
    int chunk = min(64, p1 - base);
    if (f < chunk) sidx[f] = srcSorted[base + f];
    __syncthreads();
    for (int j = 0; j < chunk; ++j)
      s += h[(size_t)sidx[j] * d + f];
    __syncthreads();
  }
  float degf = (float)(p1 - p0);
  meanb[(size_t)node * d + f] = s / fmaxf(degf, 1.0f);
}

// ------------------------------------------------------------- WMMA GEMM
// z[m,n] = sum_k mean[m,k]*wn[k,n] + sum_k h[m,k]*ws[k,n] + bias[n]
// One wave32 computes a 16x16 tile with V_WMMA_F32_16X16X4_F32 (fp32 in/acc,
// matching the fp32 reference; GEMM is only ~26 GFLOP so precision > rate).
__global__ void gnn_sage_gemm_kernel(const float* __restrict__ meanb,
                                     const float* __restrict__ h,
                                     const float* __restrict__ wn,
                                     const float* __restrict__ ws,
                                     const float* __restrict__ bias,
                                     float* __restrict__ z,
                                     int din, int dout) {
  const int lane = threadIdx.x;        // 0..31 (wave32)
  const int half = lane >> 4;
  const int r    = lane & 15;
  const int m0   = blockIdx.x * 16;
  const int n0   = (blockIdx.y * blockDim.y + threadIdx.y) * 16;

  v8f c = {};

  // pass 1: mean @ wn
  {
    const float* abase = meanb + (size_t)(m0 + r) * din + 2 * half;
    const float* bbase = wn + (size_t)(2 * half) * dout + n0 + r;
    for (int k0 = 0; k0 < din; k0 += 4) {
      v2f a = *(const v2f*)(abase + k0);
      v2f b;
      b.x = bbase[(size_t)k0 * dout];
      b.y = bbase[(size_t)(k0 + 1) * dout];
      c = __builtin_amdgcn_wmma_f32_16x16x4_f32(false, a, false, b,
                                                (short)0, c, false, false);
    }
  }
  // pass 2: h @ ws (accumulate into same C)
  {
    const float* abase = h + (size_t)(m0 + r) * din + 2 * half;
    const float* bbase = ws + (size_t)(2 * half) * dout + n0 + r;
    for (int k0 = 0; k0 < din; k0 += 4) {
      v2f a = *(const v2f*)(abase + k0);
      v2f b;
      b.x = bbase[(size_t)k0 * dout];
      b.y = bbase[(size_t)(k0 + 1) * dout];
      c = __builtin_amdgcn_wmma_f32_16x16x4_f32(false, a, false, b,
                                                (short)0, c, false, false);
    }
  }

  const float bv = bias[n0 + r];
#pragma unroll
  for (int v = 0; v < 8; ++v) {
    int row = m0 + v + 8 * half;
    z[(size_t)row * dout + n0 + r] = c[v] + bv;
  }
}

// ----------------------------------------------------------------- batchnorm
__global__ void gnn_bn_stats_kernel(const float* __restrict__ z,
                                    float* __restrict__ sums,
                                    float* __restrict__ sumsq,
                                    int N, int dout, int rowsPerBlock) {
  int col = threadIdx.x;               // blockDim.x == dout -> coalesced
  int r0 = blockIdx.x * rowsPerBlock;
  int r1 = min(r0 + rowsPerBlock, N);
  float s = 0.0f, sq = 0.0f;
  for (int row = r0; row < r1; ++row) {
    float v = z[(size_t)row * dout + col];
    s += v;
    sq += v * v;
  }
  atomicAdd(&sums[col], s);
  atomicAdd(&sumsq[col], sq);
}

__global__ void gnn_bn_finalize_kernel(const float* __restrict__ sums,
                                       const float* __restrict__ sumsq,
                                       const float* __restrict__ gamma,
                                       const float* __restrict__ beta,
                                       float* __restrict__ scale,
                                       float* __restrict__ shift,
                                       int N, int dout) {
  int j = blockIdx.x * blockDim.x + threadIdx.x;
  if (j >= dout) return;
  float invN = 1.0f / (float)N;
  float mu = sums[j] * invN;
  float var = sumsq[j] * invN - mu * mu;
  float sc = gamma[j] * rsqrtf(var + GNN_BN_EPS);
  scale[j] = sc;
  shift[j] = beta[j] - mu * sc;
}

__global__ void gnn_bn_apply_kernel(float* __restrict__ z,
                                    const float* __restrict__ scale,
                                    const float* __restrict__ shift,
                                    long long n, int colMask, int doRelu) {
  long long i = (long long)blockIdx.x * blockDim.x + threadIdx.x;
  if (i >= n) return;
  int col = (int)(i & colMask);
  float v = z[i] * scale[col] + shift[col];
  if (doRelu) v = fmaxf(v, 0.0f);
  z[i] = v;
}

// ------------------------------------------------------------------ launcher
extern "C" void kernel_launch(void* const* d_in, const int* in_sizes, int n_in,
                              void* d_out, int out_size, void* d_ws, size_t ws_size,
                              hipStream_t stream) {
  (void)in_sizes; (void)n_in; (void)out_size; (void)ws_size;

  const float* x = (const float*)d_in[0];
  const int* ei  = (const int*)d_in[1];
  const int* src = ei;                  // edge_index[0, :]
  const int* dst = ei + GNN_N_EDGES;    // edge_index[1, :]

  const float* wn[3]  = {(const float*)d_in[2],  (const float*)d_in[7],  (const float*)d_in[12]};
  const float* wsm[3] = {(const float*)d_in[3],  (const float*)d_in[8],  (const float*)d_in[13]};
  const float* bb[3]  = {(const float*)d_in[4],  (const float*)d_in[9],  (const float*)d_in[14]};
  const float* gg[3]  = {(const float*)d_in[5],  (const float*)d_in[10], (const float*)d_in[15]};
  const float* be[3]  = {(const float*)d_in[6],  (const float*)d_in[11], (const float*)d_in[16]};

  // workspace layout
  float* B0    = (float*)d_ws;                         // mean buffer [50000 x 256]
  float* B1    = B0 + (size_t)GNN_N_NODES * 256;       // h1          [50000 x 256]
  float* B2    = B1 + (size_t)GNN_N_NODES * 256;       // h2          [50000 x 256]
  float* sums  = B2 + (size_t)GNN_N_NODES * 256;       // [256]
  float* sumsq = sums + 256;                           // [256]
  float* scale = sumsq + 256;                          // [256]
  float* shift = scale + 256;                          // [256]
  int* deg       = (int*)(shift + 256);                // [50000]
  int* rowptr    = deg + GNN_N_NODES;                  // [50001]
  int* cursor    = rowptr + GNN_N_NODES + 1;           // [50000]
  int* srcSorted = cursor + GNN_N_NODES;               // [800000]

  // ---- CSR build (once per launch; graph shared by all 3 layers) ----
  gnn_zero_i32_kernel<<<(GNN_N_NODES + 255) / 256, 256, 0, stream>>>(deg, GNN_N_NODES);
  gnn_degree_kernel<<<(GNN_N_EDGES + 255) / 256, 256, 0, stream>>>(dst, deg, GNN_N_EDGES);
  gnn_scan_kernel<<<1, 1024, 0, stream>>>(deg, rowptr, GNN_N_NODES);
  gnn_copy_i32_kernel<<<(GNN_N_NODES + 255) / 256, 256, 0, stream>>>(rowptr, cursor, GNN_N_NODES);
  gnn_bucket_kernel<<<(GNN_N_EDGES + 255) / 256, 256, 0, stream>>>(src, dst, cursor,
                                                                   srcSorted, GNN_N_EDGES);

  const int dins[3]  = {128, 256, 256};
  const int douts[3] = {256, 256, 128};
  const float* hbuf[3] = {x, B1, B2};
  float* zbuf[3] = {B1, B2, (float*)d_out};

  for (int l = 0; l < 3; ++l) {
    const int din = dins[l], dout = douts[l];

    // atomic-free mean aggregation (division fused)
    gnn_gather_mean_kernel<<<GNN_N_NODES, din, 0, stream>>>(hbuf[l], rowptr,
                                                            srcSorted, B0, din);

    // z = mean @ wn + h @ ws + b via fp32 WMMA (16x16 tile per wave)
    {
      dim3 blk(32, 4);                      // 4 wave32s
      dim3 grd(GNN_N_NODES / 16, dout / 64);
      gnn_sage_gemm_kernel<<<grd, blk, 0, stream>>>(B0, hbuf[l], wn[l], wsm[l],
                                                    bb[l], zbuf[l], din, dout);
    }

    // batchnorm over node axis, fused relu for layers 0/1
    gnn_zero_f32_kernel<<<1, 256, 0, stream>>>(sums, dout);
    gnn_zero_f32_kernel<<<1, 256, 0, stream>>>(sumsq, dout);
    {
      const int ROWS = 500;
      gnn_bn_stats_kernel<<<(GNN_N_NODES + ROWS - 1) / ROWS, dout, 0, stream>>>(
          zbuf[l], sums, sumsq, GNN_N_NODES, dout, ROWS);
    }
    gnn_bn_finalize_kernel<<<1, 256, 0, stream>>>(sums, sumsq, gg[l], be[l],
                                                  scale, shift, GNN_N_NODES, dout);
    const long long nz = (long long)GNN_N_NODES * dout;
    gnn_bn_apply_kernel<<<(int)((nz + 255) / 256), 256, 0, stream>>>(
        zbuf[l], scale, shift, nz, dout - 1, (l < 2) ? 1 : 0);
  }
}